// GCN_64647847740121
// MI455X (gfx1250) — compile-verified
//
#include <hip/hip_runtime.h>

// ---------------------------------------------------------------------------
// GCN (5-layer, PyG GCNConv semantics) for MI455X / gfx1250.
// GEMMs via V_WMMA_F32_16X16X4_F32 (fp32-exact, matches reference numerics).
// Message passing via edge-parallel float4 gather + global f32 atomics.
// ---------------------------------------------------------------------------

typedef __attribute__((ext_vector_type(2))) float v2f;
typedef __attribute__((ext_vector_type(8))) float v8f;

// ---------------- degree / norm precompute ----------------
__global__ void k_deg_init(float* __restrict__ deg, int n) {
  int i = blockIdx.x * blockDim.x + threadIdx.x;
  if (i < n) deg[i] = 1.0f;  // self-loop contributes 1 to every node
}

__global__ void k_deg_count(float* __restrict__ deg, const int* __restrict__ dst, int e) {
  int i = blockIdx.x * blockDim.x + threadIdx.x;
  if (i < e) atomicAdd(&deg[dst[i]], 1.0f);
}

__global__ void k_dinv(float* __restrict__ deg, int n) {
  int i = blockIdx.x * blockDim.x + threadIdx.x;
  if (i < n) deg[i] = rsqrtf(deg[i]);  // deg >= 1 always (self-loops)
}

// ---------------- WMMA fp32 GEMM: h[N,dout] = x[N,K] * W[dout,K]^T ---------
// One wave (32 lanes) per 16x16 output tile; K consumed 4 at a time by
// v_wmma_f32_16x16x4_f32.
// A layout (16x4 f32, ISA 7.12.2): lanes 0-15 row M=lane, VGPR0=K0,VGPR1=K1;
//                                  lanes 16-31 same rows,  VGPR0=K2,VGPR1=K3.
// B layout (4x16 f32): lanes 0-15 col N=lane hold K0/K1; lanes 16-31 K2/K3.
// D layout (16x16 f32): VGPR r -> row r (lanes 0-15) / row r+8 (lanes 16-31),
//                       col = lane & 15.
__global__ void k_gemm_wmma(float* __restrict__ h, const float* __restrict__ x,
                            const float* __restrict__ w, int n, int K, int dout) {
  const int lane  = threadIdx.x;          // 0..31, full wave active (EXEC all 1s)
  const int mt    = blockIdx.x;           // tile along nodes
  const int nt    = blockIdx.y;           // tile along output features
  const int half  = lane >> 4;            // 0: K{0,1}, 1: K{2,3}
  const int l15   = lane & 15;

  int row = mt * 16 + l15;                // node row this lane loads for A
  int arow = (row < n) ? row : (n - 1);   // clamp loads (stores guarded below)
  const int col = nt * 16 + l15;          // output feature this lane loads for B

  const float* __restrict__ xr = x + (size_t)arow * K + half * 2;
  const float* __restrict__ wr = w + (size_t)col  * K + half * 2;

  v8f acc = {};
#pragma unroll 8
  for (int kk = 0; kk < K; kk += 4) {
    v2f a, b;
    a.x = xr[kk];
    a.y = xr[kk + 1];
    b.x = wr[kk];
    b.y = wr[kk + 1];
    acc = __builtin_amdgcn_wmma_f32_16x16x4_f32(
        /*neg_a=*/false, a, /*neg_b=*/false, b,
        /*c_mod=*/(short)0, acc, /*reuse_a=*/false, /*reuse_b=*/false);
  }

  const int rbase = mt * 16 + half * 8;   // rows rbase..rbase+7 for this lane
#pragma unroll
  for (int r = 0; r < 8; ++r) {
    int orow = rbase + r;
    if (orow < n) h[(size_t)orow * dout + col] = acc[r];
  }
}

// ---------------- agg init with self-loop term -----------------------------
// agg[n,f] = h[n,f] * dinv[n]^2   (the self-loop edge), replaces zero-fill.
__global__ void k_selfloop(float* __restrict__ agg, const float* __restrict__ h,
                           const float* __restrict__ dinv, int dout, long total) {
  long i = (long)blockIdx.x * blockDim.x + threadIdx.x;
  if (i < total) {
    int node = (int)(i / dout);
    float di = dinv[node];
    agg[i] = h[i] * di * di;
  }
}

// ---------------- edge-parallel scatter-add --------------------------------
// Each thread handles one float4 chunk of one edge's message.
// Consecutive threads -> consecutive chunks of the same edge (coalesced).
__global__ void k_scatter(float* __restrict__ agg, const float* __restrict__ h,
                          const float* __restrict__ dinv,
                          const int* __restrict__ src, const int* __restrict__ dst,
                          long total, int dout, int shift) {
  long gid = (long)blockIdx.x * blockDim.x + threadIdx.x;
  if (gid >= total) return;
  int edge = (int)(gid >> shift);
  int c    = (int)(gid & ((1 << shift) - 1)) << 2;   // feature offset
  int s = src[edge];
  int d = dst[edge];
  float nrm = dinv[s] * dinv[d];
  const float4 v = *(const float4*)(h + (size_t)s * dout + c);
  float* ap = agg + (size_t)d * dout + c;
  atomicAdd(ap + 0, v.x * nrm);
  atomicAdd(ap + 1, v.y * nrm);
  atomicAdd(ap + 2, v.z * nrm);
  atomicAdd(ap + 3, v.w * nrm);
}

// ---------------- bias + (optional) ReLU -----------------------------------
__global__ void k_finish(float* __restrict__ out, const float* __restrict__ agg,
                         const float* __restrict__ bias, int doutMask, long total,
                         int dorelu) {
  long i = (long)blockIdx.x * blockDim.x + threadIdx.x;
  if (i < total) {
    float v = agg[i] + bias[(int)i & doutMask];
    out[i] = dorelu ? fmaxf(v, 0.0f) : v;
  }
}

// ---------------------------------------------------------------------------
static inline size_t align256(size_t x) { return (x + 255) & ~(size_t)255; }

extern "C" void kernel_launch(void* const* d_in, const int* in_sizes, int n_in,
                              void* d_out, int out_size, void* d_ws, size_t ws_size,
                              hipStream_t stream) {
  (void)n_in; (void)out_size; (void)ws_size;

  const int DIN = 128, DH = 128, DOUT = 64, LAYERS = 5;
  const int N = in_sizes[0] / DIN;       // 50000
  const int E = in_sizes[1] / 2;         // 800000

  const float* x0   = (const float*)d_in[0];
  const int*  eidx  = (const int*)d_in[1];
  const int*  srcA  = eidx;              // edge_index[0]
  const int*  dstA  = eidx + E;          // edge_index[1]

  // workspace carve-up
  char* ws = (char*)d_ws;
  size_t off = 0;
  float* dinv = (float*)(ws + off); off += align256((size_t)N * sizeof(float));
  float* bufH = (float*)(ws + off); off += align256((size_t)N * DH * sizeof(float));
  float* bufA = (float*)(ws + off); off += align256((size_t)N * DH * sizeof(float));
  float* bufX = (float*)(ws + off); off += align256((size_t)N * DH * sizeof(float));

  const int TB = 256;

  // degree / normalization (reused by all layers)
  k_deg_init <<<(N + TB - 1) / TB, TB, 0, stream>>>(dinv, N);
  k_deg_count<<<(E + TB - 1) / TB, TB, 0, stream>>>(dinv, dstA, E);
  k_dinv     <<<(N + TB - 1) / TB, TB, 0, stream>>>(dinv, N);

  const float* xin = x0;
  for (int layer = 0; layer < LAYERS; ++layer) {
    const int K    = DIN;                       // 128 for all layers
    const int dout = (layer == LAYERS - 1) ? DOUT : DH;
    const float* W = (const float*)d_in[2 + 2 * layer];
    const float* b = (const float*)d_in[3 + 2 * layer];
    float* out = (layer == LAYERS - 1) ? (float*)d_out : bufX;

    // GEMM: one wave per 16x16 tile
    dim3 grid((N + 15) / 16, dout / 16);
    k_gemm_wmma<<<grid, 32, 0, stream>>>(bufH, xin, W, N, K, dout);

    long tot = (long)N * dout;
    k_selfloop<<<(unsigned)((tot + TB - 1) / TB), TB, 0, stream>>>(bufA, bufH, dinv, dout, tot);

    const int shift = (dout == 128) ? 5 : 4;    // float4 chunks per edge
    long sc_total = (long)E << shift;
    k_scatter<<<(unsigned)((sc_total + TB - 1) / TB), TB, 0, stream>>>(
        bufA, bufH, dinv, srcA, dstA, sc_total, dout, shift);

    k_finish<<<(unsigned)((tot + TB - 1) / TB), TB, 0, stream>>>(
        out, bufA, b, dout - 1, tot, layer < LAYERS - 1 ? 1 : 0);

    xin = bufX;
  }
}